// ImprovedSpatialInSARModel_85779086835973
// MI455X (gfx1250) — compile-verified
//
#include <hip/hip_runtime.h>
#include <math.h>

#define N_PIX   100000
#define KNB     12
#define T_LEN   365
#define T_PAD   368            // 23 tiles of 16
#define NCOEF   12             // 10 real coefficients + 2 zero pad (3 x K=4 WMMA chunks)
#define M_TILES (N_PIX / 16)   // 6250 exactly
#define T_TILES (T_PAD / 16)   // 23
#define TWO_PI_F 6.28318530717958647692f

typedef __attribute__((ext_vector_type(2))) float v2f;
typedef __attribute__((ext_vector_type(8))) float v8f;

// ---------------------------------------------------------------------------
// Kernel 1: per-pixel smoothed coefficients -> P[N][12]
//   col 0 : constant offset            (basis row 0 = 1)
//   col 1 : linear trend               (basis row 1 = t)
//   col 2+2i : amp_i * cos(phi_i')     (basis row = sin(w_i t))
//   col 3+2i : amp_i * sin(phi_i')     (basis row = cos(w_i t))
//   cols 10,11 : 0
// ---------------------------------------------------------------------------
__global__ void coeff_kernel(const float* __restrict__ off,
                             const float* __restrict__ trd,
                             const float* __restrict__ amps,
                             const float* __restrict__ phs,
                             const float* __restrict__ wgt,
                             const int*   __restrict__ idx,
                             float* __restrict__ P)
{
    int n = blockIdx.x * blockDim.x + threadIdx.x;
    if (n >= N_PIX) return;

    float a_self[4], p_self[4];
    {
        float4 a4 = reinterpret_cast<const float4*>(amps)[n];
        float4 p4 = reinterpret_cast<const float4*>(phs)[n];
        a_self[0] = a4.x; a_self[1] = a4.y; a_self[2] = a4.z; a_self[3] = a4.w;
        p_self[0] = p4.x; p_self[1] = p4.y; p_self[2] = p4.z; p_self[3] = p4.w;
    }

    float ampn[4][KNB];
    float wavg[4] = {0.f, 0.f, 0.f, 0.f};
    float asum[4] = {0.f, 0.f, 0.f, 0.f};
    float wre[4]  = {0.f, 0.f, 0.f, 0.f};
    float wim[4]  = {0.f, 0.f, 0.f, 0.f};

    #pragma unroll
    for (int k = 0; k < KNB; ++k) {
        int   j  = idx[n * KNB + k];
        float wk = wgt[n * KNB + k];
        float4 a4 = reinterpret_cast<const float4*>(amps)[j];
        float4 p4 = reinterpret_cast<const float4*>(phs)[j];
        float av[4] = {a4.x, a4.y, a4.z, a4.w};
        float pv[4] = {p4.x, p4.y, p4.z, p4.w};
        #pragma unroll
        for (int i = 0; i < 4; ++i) {
            ampn[i][k] = av[i];
            wavg[i] += av[i] * wk;
            asum[i] += av[i];
            wre[i]  += cosf(pv[i]) * wk;
            wim[i]  += sinf(pv[i]) * wk;
        }
    }

    float coeff[NCOEF];
    coeff[0]  = off[n];
    coeff[1]  = trd[n];
    coeff[10] = 0.0f;
    coeff[11] = 0.0f;

    #pragma unroll
    for (int i = 0; i < 4; ++i) {
        // amplitude smoothing (two-pass ddof=1 variance, matching jnp.var)
        float mean = asum[i] * (1.0f / (float)KNB);
        float ss = 0.0f;
        #pragma unroll
        for (int k = 0; k < KNB; ++k) {
            float d = ampn[i][k] - mean;
            ss += d * d;
        }
        float var = ss * (1.0f / (float)(KNB - 1));
        float af  = 0.25f / (1.0f + 0.1f * var);
        float amp = (1.0f - af) * a_self[i] + af * wavg[i];

        // circular phase smoothing; recover cos/sin of atan2 directly
        float cons = sqrtf(wre[i] * wre[i] + wim[i] * wim[i]);
        float afp  = 0.15f * cons;
        float mre  = (1.0f - afp) * cosf(p_self[i]) + afp * wre[i];
        float mim  = (1.0f - afp) * sinf(p_self[i]) + afp * wim[i];
        float r    = sqrtf(mre * mre + mim * mim);
        float inv  = (r > 0.0f) ? (1.0f / r) : 0.0f;
        float cph  = (r > 0.0f) ? mre * inv : 1.0f;  // atan2(0,0)=0 -> cos=1
        float sph  = mim * inv;

        coeff[2 + 2 * i] = amp * cph;   // multiplies sin(w t)
        coeff[3 + 2 * i] = amp * sph;   // multiplies cos(w t)
    }

    #pragma unroll
    for (int q = 0; q < NCOEF; ++q) P[(size_t)n * NCOEF + q] = coeff[q];
}

// ---------------------------------------------------------------------------
// Kernel 2: basis table B[12][T_PAD], zero-padded beyond T_LEN and rows 10/11
// ---------------------------------------------------------------------------
__global__ void basis_kernel(const float* __restrict__ tvec, float* __restrict__ Bb)
{
    int j = blockIdx.x * blockDim.x + threadIdx.x;
    if (j >= T_PAD) return;
    bool  valid = (j < T_LEN);
    float t = valid ? tvec[j] : 0.0f;

    Bb[0 * T_PAD + j] = valid ? 1.0f : 0.0f;
    Bb[1 * T_PAD + j] = valid ? t : 0.0f;
    const float freqs[4] = {4.0f, 2.0f, 1.0f, 0.5f};  // 1/period
    #pragma unroll
    for (int i = 0; i < 4; ++i) {
        float arg = TWO_PI_F * freqs[i] * t;
        Bb[(2 + 2 * i) * T_PAD + j] = valid ? sinf(arg) : 0.0f;
        Bb[(3 + 2 * i) * T_PAD + j] = valid ? cosf(arg) : 0.0f;
    }
    Bb[10 * T_PAD + j] = 0.0f;
    Bb[11 * T_PAD + j] = 0.0f;
}

// ---------------------------------------------------------------------------
// Kernel 3: out[N][365] = P[N][12] x B[12][368] via V_WMMA_F32_16X16X4_F32.
// One wave32 per 16-pixel tile; A operand held in registers across all 23
// time tiles. K=12 -> three chained K=4 WMMAs.
// ---------------------------------------------------------------------------
__global__ void wmma_signal_kernel(const float* __restrict__ P,
                                   const float* __restrict__ Bb,
                                   float* __restrict__ out)
{
    int gthread = blockIdx.x * blockDim.x + threadIdx.x;
    int wave = gthread >> 5;
    int lane = threadIdx.x & 31;
    if (wave >= M_TILES) return;   // uniform per wave: EXEC stays all-ones

    int m0    = wave * 16;
    int mlane = lane & 15;
    int half  = lane >> 4;   // 0 or 1
    int khalf = half * 2;    // ISA A/B layout: hi half-wave holds K+2,K+3

    // A (16x4 f32, 2 VGPRs): v0 = A[m][khalf], v1 = A[m][khalf+1]
    const float* prow = P + (size_t)(m0 + mlane) * NCOEF;
    v2f a0, a1, a2;
    a0.x = prow[0 + khalf];  a0.y = prow[1 + khalf];
    a1.x = prow[4 + khalf];  a1.y = prow[5 + khalf];
    a2.x = prow[8 + khalf];  a2.y = prow[9 + khalf];

    for (int tt = 0; tt < T_TILES; ++tt) {
        int col = tt * 16 + mlane;
        const float* bcol = Bb + col;
        // B (4x16 f32, 2 VGPRs): v0 = B[khalf][col], v1 = B[khalf+1][col]
        v2f b0, b1, b2;
        b0.x = bcol[(0 + khalf) * T_PAD];  b0.y = bcol[(1 + khalf) * T_PAD];
        b1.x = bcol[(4 + khalf) * T_PAD];  b1.y = bcol[(5 + khalf) * T_PAD];
        b2.x = bcol[(8 + khalf) * T_PAD];  b2.y = bcol[(9 + khalf) * T_PAD];

        v8f c = {};
        c = __builtin_amdgcn_wmma_f32_16x16x4_f32(false, a0, false, b0,
                                                  (short)0, c, false, false);
        c = __builtin_amdgcn_wmma_f32_16x16x4_f32(false, a1, false, b1,
                                                  (short)0, c, false, false);
        c = __builtin_amdgcn_wmma_f32_16x16x4_f32(false, a2, false, b2,
                                                  (short)0, c, false, false);

        if (col < T_LEN) {   // divergence only at stores, after WMMAs
            #pragma unroll
            for (int v = 0; v < 8; ++v) {
                int m = m0 + v + 8 * half;  // D layout: VGPR v -> row v + 8*half
                out[(size_t)m * T_LEN + col] = c[v];
            }
        }
    }
}

// ---------------------------------------------------------------------------
extern "C" void kernel_launch(void* const* d_in, const int* in_sizes, int n_in,
                              void* d_out, int out_size, void* d_ws, size_t ws_size,
                              hipStream_t stream)
{
    (void)in_sizes; (void)n_in; (void)out_size; (void)ws_size;

    const float* tvec = (const float*)d_in[0];  // time_vector        [365]
    const float* off  = (const float*)d_in[1];  // constant_offset    [N]
    const float* trd  = (const float*)d_in[2];  // linear_trend       [N]
    const float* amps = (const float*)d_in[3];  // seasonal_amplitudes[N,4]
    const float* phs  = (const float*)d_in[4];  // seasonal_phases    [N,4]
    const float* wgt  = (const float*)d_in[5];  // neighbor_weights   [N,K]
    const int*   idx  = (const int*)  d_in[6];  // neighbor_indices   [N,K]
    float* out = (float*)d_out;

    float* P  = (float*)d_ws;                   // N*12 floats (4.8 MB)
    float* Bb = P + (size_t)N_PIX * NCOEF;      // 12*368 floats (17.7 KB), 16B aligned

    basis_kernel<<<(T_PAD + 255) / 256, 256, 0, stream>>>(tvec, Bb);
    coeff_kernel<<<(N_PIX + 255) / 256, 256, 0, stream>>>(off, trd, amps, phs,
                                                          wgt, idx, P);

    int waves   = M_TILES;                      // 6250 wave32 tiles
    int threads = waves * 32;
    wmma_signal_kernel<<<(threads + 255) / 256, 256, 0, stream>>>(P, Bb, out);
}